// DeltaThetaGammaCLEVRN_16286515986783
// MI455X (gfx1250) — compile-verified
//
#include <hip/hip_runtime.h>
#include <math.h>

typedef __attribute__((ext_vector_type(2))) float v2f;
typedef __attribute__((ext_vector_type(8))) float v8f;

#define KC 152  // combined feature K: 24 ones + 64 gamma amps + 64 query feats

__device__ __forceinline__ v8f wmma_f32(v2f a, v2f b, v8f c) {
  // 8 args: (neg_a, A, neg_b, B, c_mod, C, reuse_a, reuse_b) -> v_wmma_f32_16x16x4_f32
  return __builtin_amdgcn_wmma_f32_16x16x4_f32(false, a, false, b, (short)0, c, false, false);
}

__global__ __launch_bounds__(64) void fused_osc_readout(
    const float* __restrict__ scene,   // (B,8,16)
    const float* __restrict__ query,   // (B,60)
    const float* __restrict__ scene_w, // (16,64)
    const float* __restrict__ scene_b, // (64)
    const float* __restrict__ query_w, // (60,64)
    const float* __restrict__ query_b, // (64)
    const float* __restrict__ ro_w1,   // (152,64)
    const float* __restrict__ ro_b1,   // (64)
    const float* __restrict__ ro_w2,   // (64,2)
    const float* __restrict__ ro_b2,   // (2)
    float* __restrict__ out,           // (B,2) log-softmax
    int B)
{
  // per-wave staging buffers (2 waves / block)
  __shared__ float sfA [2][16 * 16];   // scene mean tile (A for GEMM1)
  __shared__ float qA  [2][16 * 64];   // query tile, stride 64, cols 0..59 valid
  __shared__ float comb[2][16 * KC];   // combined features (A for GEMM3)
  __shared__ float hbuf[2][16 * 64];   // hidden activations

  if (blockIdx.x * 32 >= B) return;    // uniform across block
  const int w    = threadIdx.x >> 5;
  const int lane = threadIdx.x & 31;
  const int s0   = blockIdx.x * 32 + w * 16;   // this wave's 16 samples
  const int mn   = lane & 15;                  // M (A-side) or N (B/D-side)
  const int half = lane >> 4;                  // K-half for A/B, M-half for C/D

  __builtin_prefetch(scene + (size_t)s0 * 128, 0, 1);
  __builtin_prefetch(query + (size_t)s0 * 60, 0, 1);

  // ---- stage A tiles -------------------------------------------------------
  // scene mean: (16 samples x 16 feats)
  for (int idx = lane; idx < 256; idx += 32) {
    int sl = idx >> 4, f = idx & 15;
    const float* sp = scene + (size_t)(s0 + sl) * 128 + f;
    float acc = 0.f;
#pragma unroll
    for (int i = 0; i < 8; ++i) acc += sp[i * 16];
    sfA[w][idx] = acc * 0.125f;
  }
  // query tile: 16 x 60 (row stride 64)
  for (int idx = lane; idx < 16 * 60; idx += 32) {
    int sl = idx / 60, c = idx % 60;
    qA[w][sl * 64 + c] = query[(size_t)(s0 + sl) * 60 + c];
  }
  // delta+theta amplitudes stay exactly 1.0 (a*(1-a^2)==0 at a=1)
  for (int idx = lane; idx < 16 * 24; idx += 32) {
    int sl = idx / 24, c = idx % 24;
    comb[w][sl * KC + c] = 1.0f;
  }
  __syncthreads();

  // ---- GEMM1: scene_mean(16x16) @ scene_w(16x64) -> sigmoid -> 20x ODE -----
  for (int nt = 0; nt < 4; ++nt) {
    const int ncol = nt * 16 + mn;
    v8f c8 = {};
#pragma unroll
    for (int kt = 0; kt < 4; ++kt) {
      const int k0 = kt * 4 + 2 * half;
      v2f a, b;
      a.x = sfA[w][mn * 16 + k0];
      a.y = sfA[w][mn * 16 + k0 + 1];
      b.x = scene_w[k0 * 64 + ncol];
      b.y = scene_w[(k0 + 1) * 64 + ncol];
      c8 = wmma_f32(a, b, c8);
    }
    const float bias = scene_b[ncol];
#pragma unroll
    for (int r = 0; r < 8; ++r) {
      const int m = r + 8 * half;
      float x = c8[r] + bias;
      x = 1.f / (1.f + expf(-x));          // gamma_init
#pragma unroll
      for (int t = 0; t < 20; ++t)          // Stuart-Landau amplitude relaxation
        x += 0.01f * x * (1.f - x * x);
      comb[w][m * KC + 24 + ncol] = x;      // ag
    }
  }

  // ---- GEMM2: query(16x60) @ query_w(60x64) + b ----------------------------
  for (int nt = 0; nt < 4; ++nt) {
    const int ncol = nt * 16 + mn;
    v8f c8 = {};
    for (int kt = 0; kt < 15; ++kt) {
      const int k0 = kt * 4 + 2 * half;
      v2f a, b;
      a.x = qA[w][mn * 64 + k0];
      a.y = qA[w][mn * 64 + k0 + 1];
      b.x = query_w[k0 * 64 + ncol];
      b.y = query_w[(k0 + 1) * 64 + ncol];
      c8 = wmma_f32(a, b, c8);
    }
    const float bias = query_b[ncol];
#pragma unroll
    for (int r = 0; r < 8; ++r) {
      const int m = r + 8 * half;
      comb[w][m * KC + 88 + ncol] = c8[r] + bias;
    }
  }
  __syncthreads();

  // ---- GEMM3: combined(16x152) @ ro_w1(152x64) + b1, ReLU ------------------
  for (int nt = 0; nt < 4; ++nt) {
    const int ncol = nt * 16 + mn;
    v8f c8 = {};
    for (int kt = 0; kt < 38; ++kt) {
      const int k0 = kt * 4 + 2 * half;
      v2f a, b;
      a.x = comb[w][mn * KC + k0];
      a.y = comb[w][mn * KC + k0 + 1];
      b.x = ro_w1[k0 * 64 + ncol];
      b.y = ro_w1[(k0 + 1) * 64 + ncol];
      c8 = wmma_f32(a, b, c8);
    }
    const float bias = ro_b1[ncol];
#pragma unroll
    for (int r = 0; r < 8; ++r) {
      const int m = r + 8 * half;
      hbuf[w][m * 64 + ncol] = fmaxf(c8[r] + bias, 0.f);
    }
  }
  __syncthreads();

  // ---- GEMM4 (64x2) + log_softmax: VALU, one (sample,class) per lane -------
  {
    const int sl = mn, cls = half;               // 16 samples x 2 classes = 32 lanes
    float acc = ro_b2[cls];
#pragma unroll 8
    for (int j = 0; j < 64; ++j)
      acc += hbuf[w][sl * 64 + j] * ro_w2[j * 2 + cls];
    const float other = __shfl_xor(acc, 16, 32); // partner class logit
    const float mx  = fmaxf(acc, other);
    const float lse = mx + logf(expf(acc - mx) + expf(other - mx));
    out[(size_t)(s0 + sl) * 2 + cls] = acc - lse;
  }
}

extern "C" void kernel_launch(void* const* d_in, const int* in_sizes, int n_in,
                              void* d_out, int out_size, void* d_ws, size_t ws_size,
                              hipStream_t stream) {
  const float* scene   = (const float*)d_in[0];
  const float* query   = (const float*)d_in[1];
  // d_in[2..7] (phases, freqs) are provably dead for the output: amplitudes
  // decouple from phase dynamics and only amplitudes feed the readout.
  const float* scene_w = (const float*)d_in[8];
  const float* scene_b = (const float*)d_in[9];
  const float* query_w = (const float*)d_in[10];
  const float* query_b = (const float*)d_in[11];
  const float* ro_w1   = (const float*)d_in[12];
  const float* ro_b1   = (const float*)d_in[13];
  const float* ro_w2   = (const float*)d_in[14];
  const float* ro_b2   = (const float*)d_in[15];

  const int B = in_sizes[0] / 128;        // scene is (B, 8, 16)
  const int grid = (B + 31) / 32;         // 32 samples per block (2 waves x 16)
  fused_osc_readout<<<grid, 64, 0, stream>>>(
      scene, query, scene_w, scene_b, query_w, query_b,
      ro_w1, ro_b1, ro_w2, ro_b2, (float*)d_out, B);
}